// FGLGeneratorHierarchical0_82480551952947
// MI455X (gfx1250) — compile-verified
//
#include <hip/hip_runtime.h>
#include <math.h>

// ---------------------------------------------------------------------------
// FGLGeneratorHierarchical, MI455X (gfx1250, wave32).
//
// Key structural fact: the reference broadcasts z across nodes, and every
// level applies a node-shared dense W then a gather of a node-constant
// tensor -> the state is node-constant at every level. The whole model is a
// per-batch MLP chain (K=144/80/48/32/24 -> N=64/32/16/8/1) followed by a
// broadcast of one scalar per batch element over 65536 leaves.
//
// Kernel 1: one workgroup, 128 threads = 4 wave32 waves. Each wave owns a
//           16-row M-tile of the [64 x K] activations and drives
//           V_WMMA_F32_16X16X4_F32 (exact f32 -> bit-compatible-ish with the
//           f32 reference) over K in steps of 4, N tiled by 16.
// Kernel 2: broadcast v[64] over 16 MB of output with b128 stores
//           (bandwidth-bound: ~0.7us at 23.3 TB/s).
// ---------------------------------------------------------------------------

typedef __attribute__((ext_vector_type(2))) float v2f;
typedef __attribute__((ext_vector_type(8))) float v8f;

#define XS    148   // LDS X row stride (floats); 148%64=20, gcd(20,64)=4 -> 16 distinct banks over a half-wave
#define BATCH 64

// One FGL level: Y[64xN] = X[64xK] @ W[KxN] (+bias, leaky), via wmma f32 16x16x4.
// A layout (ISA 7.12.2, 32-bit A 16x4): lane L: m=L%16, VGPR0=K(k0+2*half), VGPR1=K(+1).
// B layout (column per lane, K split by half-wave, same as 16-bit B in 7.12.4).
// C/D layout: VGPR r -> rows r (lanes 0-15) and r+8 (lanes 16-31), col = lane%16.
template <int K, int N, int NT, bool LEAKY, bool FINAL>
__device__ inline void fgl_level(const float* __restrict__ Wg,
                                 const float* __restrict__ bg,
                                 float* __restrict__ Xsh,
                                 float* __restrict__ vs,
                                 int wave, int lane)
{
  const int m0   = wave << 4;
  const int half = lane >> 4;
  const int l16  = lane & 15;
  const int koff = half << 1;

  v8f acc[NT];
#pragma unroll
  for (int nt = 0; nt < NT; ++nt)
#pragma unroll
    for (int r = 0; r < 8; ++r) acc[nt][r] = 0.0f;

  const int m = m0 + l16;
#pragma unroll
  for (int k0 = 0; k0 < K; k0 += 4) {
    v2f a;
    a.x = Xsh[m * XS + k0 + koff + 0];
    a.y = Xsh[m * XS + k0 + koff + 1];
#pragma unroll
    for (int nt = 0; nt < NT; ++nt) {
      const int n  = (nt << 4) + l16;
      const int nn = (n < N) ? n : 0;            // clamp so loads stay in-bounds
      const float w0 = Wg[(k0 + koff + 0) * N + nn];
      const float w1 = Wg[(k0 + koff + 1) * N + nn];
      v2f b;
      b.x = (n < N) ? w0 : 0.0f;                 // zero-pad columns for N<16
      b.y = (n < N) ? w1 : 0.0f;
      acc[nt] = __builtin_amdgcn_wmma_f32_16x16x4_f32(
          false, a, false, b, (short)0, acc[nt], false, false);
    }
  }

  __syncthreads();  // all waves finished reading X for this level

#pragma unroll
  for (int nt = 0; nt < NT; ++nt) {
#pragma unroll
    for (int r = 0; r < 8; ++r) {
      const int row = m0 + r + (half << 3);
      const int col = (nt << 4) + l16;
      if (col < N) {
        float val = acc[nt][r] + bg[col];
        if (LEAKY) val = (val > 0.0f) ? val : 0.2f * val;
        if (FINAL) { if (col == 0) vs[row] = val; }
        else       Xsh[row * XS + col] = val;
      }
    }
  }
}

__global__ __launch_bounds__(128) void fgl_chain_kernel(
    const float* __restrict__ z,
    const int*   __restrict__ studies,
    const int*   __restrict__ tasks,
    const int*   __restrict__ contrasts,
    const float* __restrict__ esv, const float* __restrict__ esg,
    const float* __restrict__ etv, const float* __restrict__ etg,
    const float* __restrict__ ecv, const float* __restrict__ ecg,
    const float* __restrict__ fc0w, const float* __restrict__ fc0b,
    const float* __restrict__ fc1w, const float* __restrict__ fc1b,
    const float* __restrict__ fc2w, const float* __restrict__ fc2b,
    const float* __restrict__ fc3w, const float* __restrict__ fc3b,
    const float* __restrict__ fc4w, const float* __restrict__ fc4b,
    const float* __restrict__ W0, const float* __restrict__ b0,
    const float* __restrict__ W1, const float* __restrict__ b1,
    const float* __restrict__ W2, const float* __restrict__ b2,
    const float* __restrict__ W3, const float* __restrict__ b3,
    const float* __restrict__ W4, const float* __restrict__ b4,
    float* __restrict__ vout)
{
  __shared__ float Xsh[BATCH * XS];        // activations [64 x K], also scratch
  __shared__ float Cs[5 * BATCH * 16];     // per-level contents [5][64][16]
  __shared__ float vs[BATCH];

  const int t    = threadIdx.x;
  const int wave = t >> 5;
  const int lane = t & 31;

  // ---- phase 0: weight-normed embedding rows (se/te/ce) into Xsh scratch ----
  for (int idx = t; idx < 3 * BATCH * 16; idx += 128) {
    const int tab = idx >> 10;
    const int r   = idx & 1023;
    const int b   = r >> 4;
    const int c   = r & 15;
    const float* ev; const float* eg; int id;
    if (tab == 0)      { ev = esv; eg = esg; id = studies[b];   }
    else if (tab == 1) { ev = etv; eg = etg; id = tasks[b];     }
    else               { ev = ecv; eg = ecg; id = contrasts[b]; }
    float ss = 0.0f;
#pragma unroll
    for (int j = 0; j < 16; ++j) { float x = ev[id * 16 + j]; ss += x * x; }
    Xsh[idx] = eg[id] * ev[id * 16 + c] / sqrtf(ss);
  }
  __syncthreads();

  // ---- phase 1: content FCs  C_i = cat @ fc_i_w + fc_i_b ----
  for (int idx = t; idx < 5 * BATCH * 16; idx += 128) {
    const int i = idx >> 10;
    const int r = idx & 1023;
    const int b = r >> 4;
    const int c = r & 15;
    const float* fw; const float* fb; int fan;
    if (i == 0)      { fw = fc0w; fb = fc0b; fan = 16; }
    else if (i == 1) { fw = fc1w; fb = fc1b; fan = 32; }
    else if (i == 2) { fw = fc2w; fb = fc2b; fan = 48; }
    else if (i == 3) { fw = fc3w; fb = fc3b; fan = 48; }
    else             { fw = fc4w; fb = fc4b; fan = 48; }
    float acc = fb[c];
    for (int j = 0; j < fan; ++j)
      acc += Xsh[(j >> 4) * 1024 + (b << 4) + (j & 15)] * fw[j * 16 + c];
    Cs[idx] = acc;
  }
  __syncthreads();

  // ---- phase 2: build X0 = [z | C0]  (overwrites scratch) ----
  for (int idx = t; idx < BATCH * 144; idx += 128) {
    const int b = idx / 144;
    const int k = idx % 144;
    Xsh[b * XS + k] = (k < 128) ? z[b * 128 + k] : Cs[(b << 4) + (k - 128)];
  }
  __syncthreads();

  // ---- levels 0..4 (each fgl_level barriers internally before writeback) ----
  fgl_level<144, 64, 4, true, false>(W0, b0, Xsh, vs, wave, lane);
  for (int idx = t; idx < 1024; idx += 128)            // append C1 at col 64
    Xsh[(idx >> 4) * XS + 64 + (idx & 15)] = Cs[1 * 1024 + idx];
  __syncthreads();

  fgl_level<80, 32, 2, true, false>(W1, b1, Xsh, vs, wave, lane);
  for (int idx = t; idx < 1024; idx += 128)            // append C2 at col 32
    Xsh[(idx >> 4) * XS + 32 + (idx & 15)] = Cs[2 * 1024 + idx];
  __syncthreads();

  fgl_level<48, 16, 1, true, false>(W2, b2, Xsh, vs, wave, lane);
  for (int idx = t; idx < 1024; idx += 128)            // append C3 at col 16
    Xsh[(idx >> 4) * XS + 16 + (idx & 15)] = Cs[3 * 1024 + idx];
  __syncthreads();

  fgl_level<32, 8, 1, true, false>(W3, b3, Xsh, vs, wave, lane);
  for (int idx = t; idx < 1024; idx += 128)            // append C4 at col 8
    Xsh[(idx >> 4) * XS + 8 + (idx & 15)] = Cs[4 * 1024 + idx];
  __syncthreads();

  fgl_level<24, 1, 1, false, true>(W4, b4, Xsh, vs, wave, lane);
  __syncthreads();

  if (t < BATCH) vout[t] = vs[t];
}

// Broadcast v[b] over 65536 leaves per batch element: 1M x b128 stores.
__global__ __launch_bounds__(256) void fgl_bcast_kernel(
    const float* __restrict__ v, float* __restrict__ out)
{
  const int id = blockIdx.x * 256 + threadIdx.x;  // 0 .. 1048575 (float4 units)
  const int b  = id >> 14;                        // 16384 float4 per batch elem
  const float val = v[b];
  const float4 f4 = make_float4(val, val, val, val);
  reinterpret_cast<float4*>(out)[id] = f4;
}

extern "C" void kernel_launch(void* const* d_in, const int* in_sizes, int n_in,
                              void* d_out, int out_size, void* d_ws, size_t ws_size,
                              hipStream_t stream) {
  const float* z         = (const float*)d_in[0];
  const int*   studies   = (const int*)d_in[1];
  const int*   tasks     = (const int*)d_in[2];
  const int*   contrasts = (const int*)d_in[3];
  // d_in[4..8] = p0..p4 parent indices: provably irrelevant (node-constant state)
  const float* esv = (const float*)d_in[9];
  const float* esg = (const float*)d_in[10];
  const float* etv = (const float*)d_in[11];
  const float* etg = (const float*)d_in[12];
  const float* ecv = (const float*)d_in[13];
  const float* ecg = (const float*)d_in[14];
  const float* fc0w = (const float*)d_in[15]; const float* fc0b = (const float*)d_in[16];
  const float* fc1w = (const float*)d_in[17]; const float* fc1b = (const float*)d_in[18];
  const float* fc2w = (const float*)d_in[19]; const float* fc2b = (const float*)d_in[20];
  const float* fc3w = (const float*)d_in[21]; const float* fc3b = (const float*)d_in[22];
  const float* fc4w = (const float*)d_in[23]; const float* fc4b = (const float*)d_in[24];
  const float* W0 = (const float*)d_in[25]; const float* b0 = (const float*)d_in[26];
  const float* W1 = (const float*)d_in[27]; const float* b1 = (const float*)d_in[28];
  const float* W2 = (const float*)d_in[29]; const float* b2 = (const float*)d_in[30];
  const float* W3 = (const float*)d_in[31]; const float* b3 = (const float*)d_in[32];
  const float* W4 = (const float*)d_in[33]; const float* b4 = (const float*)d_in[34];

  float* v   = (float*)d_ws;    // 64 floats of scratch
  float* out = (float*)d_out;   // [64, 65536, 1] f32

  fgl_chain_kernel<<<1, 128, 0, stream>>>(
      z, studies, tasks, contrasts,
      esv, esg, etv, etg, ecv, ecg,
      fc0w, fc0b, fc1w, fc1b, fc2w, fc2b, fc3w, fc3b, fc4w, fc4b,
      W0, b0, W1, b1, W2, b2, W3, b3, W4, b4, v);

  // 64 * 65536 floats = 1,048,576 float4 stores
  fgl_bcast_kernel<<<4096, 256, 0, stream>>>(v, out);
}